// EndpointRegressor_66099546685672
// MI455X (gfx1250) — compile-verified
//
#include <hip/hip_runtime.h>

#define HID 160
#define JKC 320
#define NH 4
#define HD 40
#define GG 32
#define PSTRIDE 352                    // LDS row stride (floats); 352%64==32 -> conflict-free b64
#define LDS_BYTES (82 * PSTRIDE * 4)   // 80 k-pair rows + 2 pad rows for pipeline tail

typedef __attribute__((ext_vector_type(2))) float v2f;
typedef __attribute__((ext_vector_type(8))) float v8f;

// ---------- helpers ----------
__device__ __forceinline__ unsigned fkey(float f) {
  unsigned u = __float_as_uint(f);
  return (u & 0x80000000u) ? ~u : (u | 0x80000000u);
}
__device__ __forceinline__ float funkey(unsigned k) {
  return (k & 0x80000000u) ? __uint_as_float(k ^ 0x80000000u) : __uint_as_float(~k);
}
__device__ __forceinline__ void atomAddF(float* p, float v) {
  __hip_atomic_fetch_add(p, v, __ATOMIC_RELAXED, __HIP_MEMORY_SCOPE_AGENT);
}

__global__ void k_zero(float* __restrict__ p, long long n) {
  long long i = (long long)blockIdx.x * blockDim.x + threadIdx.x;
  long long stride = (long long)gridDim.x * blockDim.x;
  for (; i < n; i += stride) p[i] = 0.f;
}

// ---------- input projection: h = x @ W_in + b_in  (K=5) ----------
__global__ void k_gemm_in(const float* __restrict__ x, const float* __restrict__ W,
                          const float* __restrict__ b, float* __restrict__ h, int N) {
  long long t = (long long)blockIdx.x * blockDim.x + threadIdx.x;
  if (t >= (long long)N * HID) return;
  int n = (int)(t / HID), c = (int)(t % HID);
  float s = b[c];
#pragma unroll
  for (int j = 0; j < 5; ++j) s += x[n * 5 + j] * W[j * HID + c];
  h[t] = s;
}

// ---------- LDS-staged, double-buffered WMMA f32 GEMM ----------
// Y[M,OC] = act(X[M,K] @ W[K,OC] + b); 256 threads = 8 waves, one 16-row tile/wave.
// W staged per 160-k-row chunk into paired LDS layout: one ds_load_b64 per B fragment.
// Inner loop software-pipelined: WMMAs for step k consume bank A while ds loads for
// step k+4 fill bank B, so the DS wait is overlapped instead of blocking per-WMMA.
__global__ __launch_bounds__(256)
void k_wmma_gemm(const float* __restrict__ X, const float* __restrict__ W,
                 const float* __restrict__ b, float* __restrict__ Y,
                 int M, int K, int OCtot, int relu) {
  extern __shared__ float lw[];  // 82 * PSTRIDE floats
  const int tid = (int)threadIdx.x;
  const int lane = tid & 31;
  const int wave = tid >> 5;
  const int half = lane >> 4;
  const int idx = lane & 15;
  const int colBase = (int)blockIdx.y * 160;
  const int row0 = ((int)blockIdx.x * 8 + wave) << 4;
  const bool active = (row0 < M);

  v8f acc[10];
#pragma unroll
  for (int t = 0; t < 10; ++t) {
    float bb = b[colBase + t * 16 + idx];
#pragma unroll
    for (int r = 0; r < 8; ++r) acc[t][r] = bb;
  }

  const float* __restrict__ Xr = X + (unsigned)((active ? row0 + idx : idx) * K);
  const int lcol = idx * 2;

  for (int ks = 0; ks < K; ks += 160) {
    __syncthreads();  // previous-stage compute done before LDS overwrite
    // stage 160 k-rows x 160 cols of W into paired LDS layout
    for (int i = tid; i < 160 * 160; i += 256) {
      int kk = i / 160, c = i - kk * 160;
      float w = W[(unsigned)((ks + kk) * OCtot) + colBase + c];
      lw[(kk >> 1) * PSTRIDE + c * 2 + (kk & 1)] = w;
    }
    __syncthreads();

    if (active) {
      v2f b0[10], b1[10];
      {
        const int lb = half * PSTRIDE + lcol;  // pairs for k0=0
#pragma unroll
        for (int t = 0; t < 10; ++t) b0[t] = *(const v2f*)&lw[lb + t * 32];
      }
      for (int k0 = 0; k0 < 160; k0 += 8) {
        v2f a, a2;
        a.x = Xr[ks + k0 + (half << 1)];
        a.y = Xr[ks + k0 + (half << 1) + 1];
        a2.x = Xr[ks + k0 + 4 + (half << 1)];
        a2.y = Xr[ks + k0 + 4 + (half << 1) + 1];
        {
          const int lb = (((k0 + 4) >> 1) + half) * PSTRIDE + lcol;
#pragma unroll
          for (int t = 0; t < 10; ++t) b1[t] = *(const v2f*)&lw[lb + t * 32];
        }
#pragma unroll
        for (int t = 0; t < 10; ++t)
          acc[t] = __builtin_amdgcn_wmma_f32_16x16x4_f32(false, a, false, b0[t],
                                                         (short)0, acc[t], false, false);
        {
          // last iteration reads the (unused) pad rows 80/81 -- allocated, never consumed
          const int lb = (((k0 + 8) >> 1) + half) * PSTRIDE + lcol;
#pragma unroll
          for (int t = 0; t < 10; ++t) b0[t] = *(const v2f*)&lw[lb + t * 32];
        }
#pragma unroll
        for (int t = 0; t < 10; ++t)
          acc[t] = __builtin_amdgcn_wmma_f32_16x16x4_f32(false, a2, false, b1[t],
                                                         (short)0, acc[t], false, false);
      }
    }
  }

  if (active) {
#pragma unroll
    for (int t = 0; t < 10; ++t) {
      const int col = colBase + t * 16 + idx;
#pragma unroll
      for (int rr = 0; rr < 8; ++rr) {
        float v = acc[t][rr];
        if (relu) v = fmaxf(v, 0.f);
        Y[(unsigned)((row0 + rr + (half << 3)) * OCtot) + col] = v;
      }
    }
  }
}

// ---------- edge feature helper: e = edge_attr @ We + be, one column ----------
__device__ __forceinline__ float edge_feat(const float* __restrict__ We,
                                           const float* __restrict__ be,
                                           float a0, float a1, float a2, float a3, int c) {
  return be[c] + a0 * We[c] + a1 * We[HID + c] + a2 * We[2 * HID + c] + a3 * We[3 * HID + c];
}

// ---------- alpha[e,h] = q[dst]·(k[src]+e)/sqrt(D), atomicMax per (dst,h) ----------
__global__ void k_edge_alpha(const long long* __restrict__ ei, const float* __restrict__ q,
                             const float* __restrict__ kk, const float* __restrict__ ea,
                             const float* __restrict__ We, const float* __restrict__ be,
                             float* __restrict__ alpha, unsigned* __restrict__ amaxk, int E) {
  int t = blockIdx.x * blockDim.x + threadIdx.x;
  if (t >= E * NH) return;
  int e = t >> 2, h = t & 3;
  int src = (int)ei[e];
  int dst = (int)ei[E + e];
  float a0 = ea[e * 4 + 0], a1 = ea[e * 4 + 1], a2 = ea[e * 4 + 2], a3 = ea[e * 4 + 3];
  const float* qp = q + (long long)dst * HID + h * HD;
  const float* kp = kk + (long long)src * HID + h * HD;
  float s = 0.f;
#pragma unroll 8
  for (int i = 0; i < HD; ++i) {
    int c = h * HD + i;
    float ev = edge_feat(We, be, a0, a1, a2, a3, c);
    s += qp[i] * (kp[i] + ev);
  }
  s *= 0.15811388300841898f;  // 1/sqrt(40)
  alpha[t] = s;
  atomicMax(&amaxk[dst * NH + h], fkey(s));
}

// ---------- ex = exp(alpha - amax[dst]), denom += ex ----------
__global__ void k_edge_ex(const long long* __restrict__ ei, const unsigned* __restrict__ amaxk,
                          float* __restrict__ alpha, float* __restrict__ denom, int E) {
  int t = blockIdx.x * blockDim.x + threadIdx.x;
  if (t >= E * NH) return;
  int e = t >> 2, h = t & 3;
  int dst = (int)ei[E + e];
  float ex = __expf(alpha[t] - funkey(amaxk[dst * NH + h]));
  alpha[t] = ex;
  atomAddF(&denom[dst * NH + h], ex);
}

// ---------- agg[dst] += (v[src]+e) * ex/denom[dst] ----------
__global__ void k_edge_scatter(const long long* __restrict__ ei, const float* __restrict__ vv,
                               const float* __restrict__ ea, const float* __restrict__ We,
                               const float* __restrict__ be, const float* __restrict__ alpha,
                               const float* __restrict__ denom, float* __restrict__ agg, int E) {
  int t = blockIdx.x * blockDim.x + threadIdx.x;
  if (t >= E * NH) return;
  int e = t >> 2, h = t & 3;
  int src = (int)ei[e];
  int dst = (int)ei[E + e];
  float a = alpha[t] / denom[dst * NH + h];
  float a0 = ea[e * 4 + 0], a1 = ea[e * 4 + 1], a2 = ea[e * 4 + 2], a3 = ea[e * 4 + 3];
  const float* vp = vv + (long long)src * HID + h * HD;
  float* op = agg + (long long)dst * HID + h * HD;
#pragma unroll 8
  for (int i = 0; i < HD; ++i) {
    int c = h * HD + i;
    float ev = edge_feat(We, be, a0, a1, a2, a3, c);
    atomAddF(&op[i], (vp[i] + ev) * a);
  }
}

// ---------- beta gate + JK concat write (wave per node) ----------
__global__ __launch_bounds__(32)
void k_beta(const float* __restrict__ agg, const float* __restrict__ r,
            const float* __restrict__ Wb, float* __restrict__ h,
            float* __restrict__ xc, int layerOff) {
  int n = blockIdx.x;
  int lane = threadIdx.x;
  float s = 0.f;
  for (int c = lane; c < HID; c += 32) {
    float o = agg[(long long)n * HID + c], rr = r[(long long)n * HID + c];
    s += o * Wb[c] + rr * Wb[HID + c] + (o - rr) * Wb[2 * HID + c];
  }
#pragma unroll
  for (int off = 16; off > 0; off >>= 1) s += __shfl_xor(s, off, 32);
  float beta = 1.f / (1.f + __expf(-s));
  for (int c = lane; c < HID; c += 32) {
    float o = agg[(long long)n * HID + c], rr = r[(long long)n * HID + c];
    float hn = beta * rr + (1.f - beta) * o;
    h[(long long)n * HID + c] = hn;
    xc[(long long)n * JKC + layerOff + c] = hn;
  }
}

// ---------- gate scalar = g1 @ Wg2 + bg2 (wave per node) + graph max ----------
__global__ __launch_bounds__(32)
void k_gate(const float* __restrict__ g1, const float* __restrict__ Wg2,
            const float* __restrict__ bg2, const long long* __restrict__ batch,
            float* __restrict__ gate, unsigned* __restrict__ gmaxk) {
  int n = blockIdx.x;
  int lane = threadIdx.x;
  float s = 0.f;
  for (int c = lane; c < HID; c += 32) s += g1[(long long)n * HID + c] * Wg2[c];
#pragma unroll
  for (int off = 16; off > 0; off >>= 1) s += __shfl_xor(s, off, 32);
  s += bg2[0];
  if (lane == 0) {
    gate[n] = s;
    atomicMax(&gmaxk[(int)batch[n]], fkey(s));
  }
}

__global__ void k_gate_ex(const long long* __restrict__ batch, const unsigned* __restrict__ gmaxk,
                          float* __restrict__ gate, float* __restrict__ gden, int N) {
  int n = blockIdx.x * blockDim.x + threadIdx.x;
  if (n >= N) return;
  int b = (int)batch[n];
  float ge = __expf(gate[n] - funkey(gmaxk[b]));
  gate[n] = ge;
  atomAddF(&gden[b], ge);
}

__global__ void k_pool(const long long* __restrict__ batch, const float* __restrict__ gate,
                       const float* __restrict__ gden, const float* __restrict__ xc,
                       float* __restrict__ pooled, long long total) {
  long long t = (long long)blockIdx.x * blockDim.x + threadIdx.x;
  if (t >= total) return;
  int n = (int)(t / JKC), c = (int)(t % JKC);
  int b = (int)batch[n];
  atomAddF(&pooled[b * JKC + c], (gate[n] / gden[b]) * xc[t]);
}

__global__ void k_final(const float* __restrict__ hidden, const float* __restrict__ Wh2,
                        const float* __restrict__ bh2, float* __restrict__ out) {
  int t = blockIdx.x * blockDim.x + threadIdx.x;
  if (t >= GG * 6) return;
  int g = t / 6, o = t % 6;
  float s = bh2[o];
  for (int k = 0; k < JKC; ++k) s += hidden[g * JKC + k] * Wh2[k * 6 + o];
  out[t] = s;
}

// ---------- host launch ----------
extern "C" void kernel_launch(void* const* d_in, const int* in_sizes, int n_in,
                              void* d_out, int out_size, void* d_ws, size_t ws_size,
                              hipStream_t stream) {
  const float* x = (const float*)d_in[0];
  const long long* ei = (const long long*)d_in[1];
  const float* ea = (const float*)d_in[2];
  const long long* batch = (const long long*)d_in[3];
  const float* W_in = (const float*)d_in[4];
  const float* b_in = (const float*)d_in[5];
  const float* Wq = (const float*)d_in[6];
  const float* bq = (const float*)d_in[7];
  const float* Wk = (const float*)d_in[8];
  const float* bk = (const float*)d_in[9];
  const float* Wv = (const float*)d_in[10];
  const float* bv = (const float*)d_in[11];
  const float* We = (const float*)d_in[12];
  const float* be = (const float*)d_in[13];
  const float* Wsk = (const float*)d_in[14];
  const float* bsk = (const float*)d_in[15];
  const float* Wbeta = (const float*)d_in[16];
  const float* Wg1 = (const float*)d_in[17];
  const float* bg1 = (const float*)d_in[18];
  const float* Wg2 = (const float*)d_in[19];
  const float* bg2 = (const float*)d_in[20];
  const float* Wh1 = (const float*)d_in[21];
  const float* bh1 = (const float*)d_in[22];
  const float* Wh2 = (const float*)d_in[23];
  const float* bh2 = (const float*)d_in[24];

  const int N = in_sizes[0] / 5;
  const int E = in_sizes[1] / 2;

  float* p = (float*)d_ws;
  float* h = p;      p += (long long)N * HID;
  float* q = p;      p += (long long)N * HID;
  float* kbuf = p;   p += (long long)N * HID;
  float* vbuf = p;   p += (long long)N * HID;
  float* rbuf = p;   p += (long long)N * HID;
  float* agg = p;    p += (long long)N * HID;
  float* xc = p;     p += (long long)N * JKC;
  float* g1 = p;     p += (long long)N * HID;
  float* alpha = p;  p += (long long)E * NH;
  unsigned* amaxk = (unsigned*)p; p += (long long)N * NH;
  float* denom = p;  p += (long long)N * NH;
  float* gate = p;   p += N;
  unsigned* gmaxk = (unsigned*)p; p += GG;
  float* gden = p;   p += GG;
  float* pooled = p; p += GG * JKC;
  float* hidden = p; p += GG * JKC;

  {
    long long tot = (long long)N * HID;
    k_gemm_in<<<(int)((tot + 255) / 256), 256, 0, stream>>>(x, W_in, b_in, h, N);
  }

  const int rowTiles = N / 16;                 // 3125
  const int gemmBlocks = (rowTiles + 7) / 8;   // 391
  const long long eht = (long long)E * NH;
  const int ebl = (int)((eht + 255) / 256);

  for (int l = 0; l < 2; ++l) {
    const float* Wq_l = Wq + (long long)l * HID * HID;
    const float* Wk_l = Wk + (long long)l * HID * HID;
    const float* Wv_l = Wv + (long long)l * HID * HID;
    const float* Ws_l = Wsk + (long long)l * HID * HID;
    const float* We_l = We + (long long)l * 4 * HID;
    k_wmma_gemm<<<dim3(gemmBlocks, 1), 256, LDS_BYTES, stream>>>(h, Wq_l, bq + l * HID, q, N, HID, HID, 0);
    k_wmma_gemm<<<dim3(gemmBlocks, 1), 256, LDS_BYTES, stream>>>(h, Wk_l, bk + l * HID, kbuf, N, HID, HID, 0);
    k_wmma_gemm<<<dim3(gemmBlocks, 1), 256, LDS_BYTES, stream>>>(h, Wv_l, bv + l * HID, vbuf, N, HID, HID, 0);
    k_wmma_gemm<<<dim3(gemmBlocks, 1), 256, LDS_BYTES, stream>>>(h, Ws_l, bsk + l * HID, rbuf, N, HID, HID, 0);

    k_zero<<<256, 256, 0, stream>>>((float*)amaxk, (long long)N * NH);
    k_zero<<<256, 256, 0, stream>>>(denom, (long long)N * NH);
    k_zero<<<2048, 256, 0, stream>>>(agg, (long long)N * HID);

    k_edge_alpha<<<ebl, 256, 0, stream>>>(ei, q, kbuf, ea, We_l, be + l * HID, alpha, amaxk, E);
    k_edge_ex<<<ebl, 256, 0, stream>>>(ei, amaxk, alpha, denom, E);
    k_edge_scatter<<<ebl, 256, 0, stream>>>(ei, vbuf, ea, We_l, be + l * HID, alpha, denom, agg, E);

    k_beta<<<N, 32, 0, stream>>>(agg, rbuf, Wbeta + l * 3 * HID, h, xc, l * HID);
  }

  // attentional aggregation over graphs
  k_wmma_gemm<<<dim3(gemmBlocks, 1), 256, LDS_BYTES, stream>>>(xc, Wg1, bg1, g1, N, JKC, HID, 1);
  k_zero<<<1, 64, 0, stream>>>((float*)gmaxk, GG);
  k_zero<<<1, 64, 0, stream>>>(gden, GG);
  k_zero<<<64, 256, 0, stream>>>(pooled, GG * JKC);
  k_gate<<<N, 32, 0, stream>>>(g1, Wg2, bg2, batch, gate, gmaxk);
  k_gate_ex<<<(N + 255) / 256, 256, 0, stream>>>(batch, gmaxk, gate, gden, N);
  long long pt = (long long)N * JKC;
  k_pool<<<(int)((pt + 255) / 256), 256, 0, stream>>>(batch, gate, gden, xc, pooled, pt);

  // head MLP: [32,320] @ [320,320] (relu) then [32,320] @ [320,6]
  k_wmma_gemm<<<dim3(1, 2), 256, LDS_BYTES, stream>>>(pooled, Wh1, bh1, hidden, GG, JKC, JKC, 1);
  k_final<<<1, 256, 0, stream>>>(hidden, Wh2, bh2, (float*)d_out);
}